// BktModel_50294067036251
// MI455X (gfx1250) — compile-verified
//
#include <hip/hip_runtime.h>
#include <hip/hip_bf16.h>

// Problem constants (from reference): B=512, T=500, K=64, D=64
#define BB 512
#define TT 500
#define KK 64
#define DD 64

typedef __attribute__((ext_vector_type(2))) float v2f;
typedef __attribute__((ext_vector_type(8))) float v8f;

__device__ __forceinline__ float lse2(float a, float b) {
    float m = fmaxf(a, b);
    float d = fminf(a, b) - m;
    return m + __logf(1.0f + __expf(d));
}

// ---------------------------------------------------------------------------
// Kernel 1: decode one-hot chain [B,T,K] f32 -> packed byte (k | y<<7), stored
// time-major packed[t*B + b]. 8 lanes cooperate per row (64 floats), coalesced
// float4 loads, shuffle-max reduction of the found index.
// ---------------------------------------------------------------------------
__global__ __launch_bounds__(256) void bkt_decode_kernel(
    const float* __restrict__ chain, const int* __restrict__ corr,
    unsigned char* __restrict__ packed) {
    int tid = blockIdx.x * blockDim.x + threadIdx.x;
    int row = tid >> 3;           // (b*T + t)
    int sub = tid & 7;
    if (row >= BB * TT) return;

    const float4* p = (const float4*)(chain + (size_t)row * KK + sub * 8);
    float4 v0 = p[0];
    float4 v1 = p[1];
    int loc = -1;
    int base = sub * 8;
    if (v0.x > 0.5f) loc = base + 0;
    if (v0.y > 0.5f) loc = base + 1;
    if (v0.z > 0.5f) loc = base + 2;
    if (v0.w > 0.5f) loc = base + 3;
    if (v1.x > 0.5f) loc = base + 4;
    if (v1.y > 0.5f) loc = base + 5;
    if (v1.z > 0.5f) loc = base + 6;
    if (v1.w > 0.5f) loc = base + 7;

    // max-reduce across the 8-lane group (exactly one lane found the 1.0)
    for (int off = 1; off < 8; off <<= 1)
        loc = max(loc, __shfl_xor(loc, off, 8));

    if (sub == 0) {
        int y = corr[row];                 // corr flat [B,T] == row order
        int b = row / TT;
        int t = row - b * TT;
        packed[(size_t)t * BB + b] = (unsigned char)((loc & 63) | (y << 7));
    }
}

// ---------------------------------------------------------------------------
// Kernel 2: parameter setup.
// Phase 0: stage fw (5x64) into a zero-padded 16x64 LDS tile so the WMMA
//          K-loop is branch-free (no predicated loads, uniform EXEC).
// Phase 1 (4 waves, WMMA f32 16x16x4): l[64x16] = embd[64,64] @ fw.T[64,5->16]
// Phase 2 (64 threads): per-k log-softmax tables -> params[k*12 + ...]
//   [0..3] log_obs[s][o], [4..7] log_t[t'][s], [8..9] log_init[s]
// ---------------------------------------------------------------------------
__global__ __launch_bounds__(128) void bkt_setup_kernel(
    const float* __restrict__ embd,  // [K, D]
    const float* __restrict__ fw,    // [5, D]
    const float* __restrict__ fb,    // [5]
    float* __restrict__ params) {    // [K, 12]
    __shared__ float fwsh[16 * DD];  // zero-padded fw (rows 5..15 = 0)
    __shared__ float lsh[KK * 16];

    // Phase 0: fill padded B operand in LDS (one-shot, 8 elems per thread)
    for (int i = threadIdx.x; i < 16 * DD; i += 128)
        fwsh[i] = (i < 5 * DD) ? fw[i] : 0.0f;
    __syncthreads();

    int lane = threadIdx.x & 31;
    int wave = threadIdx.x >> 5;     // 0..3 -> M tile
    int half = lane >> 4;            // ISA: lanes 16-31 hold K=2,3 (A/B rows 2,3)
    int l15  = lane & 15;

    v8f c = {};
    for (int kk = 0; kk < 16; ++kk) {
        int kbase = kk * 4 + 2 * half;      // K = 2*laneHalf + vgpr
        int arow  = wave * 16 + l15;        // A: M = lane (mod 16)
        v2f a, b;
        a.x = embd[arow * DD + kbase + 0];
        a.y = embd[arow * DD + kbase + 1];
        int n = l15;                        // B: N = lane (mod 16); B = fw.T
        b.x = fwsh[n * DD + kbase + 0];
        b.y = fwsh[n * DD + kbase + 1];
        // D = A(16x4,f32) * B(4x16,f32) + C ; 8-arg f32 pattern
        c = __builtin_amdgcn_wmma_f32_16x16x4_f32(
                false, a, false, b, (short)0, c, false, false);
    }
    // C/D layout: VGPR r -> M = r + 8*laneHalf, N = lane&15
    for (int r = 0; r < 8; ++r) {
        int m = wave * 16 + r + 8 * half;
        int n = l15;
        lsh[m * 16 + n] = c[r] + ((n < 5) ? fb[n] : 0.0f);
    }
    __syncthreads();

    if (threadIdx.x < KK) {
        int k = threadIdx.x;
        float l0 = lsh[k * 16 + 0];
        float l1 = lsh[k * 16 + 1];
        float l2 = lsh[k * 16 + 2];
        float l3 = lsh[k * 16 + 3];
        float l4 = lsh[k * 16 + 4];

        // obs_logits: s=0 row (-l2, l2); s=1 row (l3, -l3); softmax over o
        float z0 = lse2(-l2, l2);
        float z1 = lse2(l3, -l3);
        float lo00 = -l2 - z0, lo01 = l2 - z0;
        float lo10 =  l3 - z1, lo11 = -l3 - z1;
        // trans_logits: [[-l0, l1],[l0, -l1]] (target,source); softmax over target
        float zt0 = lse2(-l0, l0);   // source column s=0: (-l0, l0)
        float zt1 = lse2(l1, -l1);   // source column s=1: (l1, -l1)
        float lt00 = -l0 - zt0, lt10 = l0 - zt0;   // lt[t'][0]
        float lt01 =  l1 - zt1, lt11 = -l1 - zt1;  // lt[t'][1]
        // init_logits: (-l4, l4); softmax
        float zi = lse2(-l4, l4);
        float li0 = -l4 - zi, li1 = l4 - zi;

        float* p = params + k * 12;
        p[0] = lo00; p[1] = lo01; p[2] = lo10; p[3] = lo11;
        p[4] = lt00; p[5] = lt01; p[6] = lt10; p[7] = lt11;
        p[8] = li0;  p[9] = li1;  p[10] = 0.0f; p[11] = 0.0f;
    }
}

// ---------------------------------------------------------------------------
// Kernel 3: the sequential HMM filter scan. One lane per batch chain,
// single-wave workgroups (16 blocks -> 16 WGPs). alpha state in LDS with a
// 130-float row pitch (130 % 64 != 0 -> conflict-free per-lane rows).
// ---------------------------------------------------------------------------
__global__ __launch_bounds__(32) void bkt_scan_kernel(
    const unsigned char* __restrict__ packed,  // [T, B]
    const float* __restrict__ params,          // [K, 12]
    float* __restrict__ out) {                 // [B, T, 2]
    __shared__ float pp[KK * 12];
    __shared__ float alpha[32 * 130];

    int lane = threadIdx.x;
    for (int i = lane; i < KK * 12; i += 32) pp[i] = params[i];
    __syncthreads();

    int gb = blockIdx.x * 32 + lane;
    float* al = alpha + lane * 130;
    #pragma unroll 4
    for (int k = 0; k < KK; ++k) {
        al[2 * k + 0] = pp[k * 12 + 8];
        al[2 * k + 1] = pp[k * 12 + 9];
    }
    __syncthreads();

    float* outp = out + (size_t)gb * TT * 2;
    for (int t = 0; t < TT; ++t) {
        if (t + 64 < TT)
            __builtin_prefetch(&packed[(size_t)(t + 64) * BB + gb], 0, 1);

        unsigned int pk = packed[(size_t)t * BB + gb];
        int k = (int)(pk & 63u);
        int y = (int)(pk >> 7);

        const float* q = pp + k * 12;
        float lo00 = q[0], lo01 = q[1], lo10 = q[2], lo11 = q[3];
        float t00  = q[4], t01  = q[5], t10  = q[6], t11  = q[7];
        float a20 = al[2 * k + 0];
        float a21 = al[2 * k + 1];

        // predictive log p(y_t): lp[o] = lse_s(log_obs[s][o] + alpha[s])
        float lp0 = lse2(lo00 + a20, lo10 + a21);
        float lp1 = lse2(lo01 + a20, lo11 + a21);
        float Z   = lse2(lp0, lp1);
        float2 o2;
        o2.x = lp0 - Z;
        o2.y = lp1 - Z;
        *(float2*)(outp + t * 2) = o2;

        // filtering update with observed y
        float ly0 = y ? lo01 : lo00;
        float ly1 = y ? lo11 : lo10;
        float u0 = ly0 + a20;
        float u1 = ly1 + a21;
        float a30 = lse2(u0 + t00, u1 + t01);
        float a31 = lse2(u0 + t10, u1 + t11);
        al[2 * k + 0] = a30;
        al[2 * k + 1] = a31;
    }
}

// ---------------------------------------------------------------------------
extern "C" void kernel_launch(void* const* d_in, const int* in_sizes, int n_in,
                              void* d_out, int out_size, void* d_ws, size_t ws_size,
                              hipStream_t stream) {
    const int*   corr  = (const int*)d_in[0];     // [B,T] int32
    const float* chain = (const float*)d_in[1];   // [B,T,K] f32 one-hot
    const float* embd  = (const float*)d_in[2];   // [K,D]
    const float* fw    = (const float*)d_in[3];   // [5,D]
    const float* fb    = (const float*)d_in[4];   // [5]
    float* out = (float*)d_out;                   // [B,T,2]

    float* params = (float*)d_ws;                                  // 64*12 f32
    unsigned char* packed = (unsigned char*)d_ws + 4096;           // T*B bytes

    // 1) decode one-hot -> packed indices (bandwidth pass over 65 MB)
    {
        int threads = BB * TT * 8;
        int blocks = (threads + 255) / 256;
        bkt_decode_kernel<<<blocks, 256, 0, stream>>>(chain, corr, packed);
    }
    // 2) WMMA parameter projection + log-softmax tables
    bkt_setup_kernel<<<1, 128, 0, stream>>>(embd, fw, fb, params);
    // 3) latency-bound 500-step filter scan, one lane per chain
    bkt_scan_kernel<<<BB / 32, 32, 0, stream>>>(packed, params, out);
}